// REConv_53704271069638
// MI455X (gfx1250) — compile-verified
//
#include <hip/hip_runtime.h>
#include <hip/hip_bf16.h>

typedef __attribute__((ext_vector_type(2))) float v2f;
typedef __attribute__((ext_vector_type(8))) float v8f;

#define F_DIM 64
#define WPAD  65   // weight LDS row stride (bank spread)
#define FPAD  68   // feat LDS row stride (float4-aligned, bank spread)

// ---------------------------------------------------------------------------
// K0: zero degree counters and the output accumulator (d_out is poisoned).
// ---------------------------------------------------------------------------
__global__ void zero_kernel(int* __restrict__ deg, long long degN,
                            float* __restrict__ out, long long outN) {
    long long i = (long long)blockIdx.x * blockDim.x + threadIdx.x;
    long long stride = (long long)gridDim.x * blockDim.x;
    for (long long j = i; j < degN; j += stride) deg[j] = 0;
    for (long long j = i; j < outN; j += stride) out[j] = 0.0f;
}

// ---------------------------------------------------------------------------
// K1: out-degree / in-degree histograms over the edge list.
// ---------------------------------------------------------------------------
__global__ void degree_kernel(const int* __restrict__ src, const int* __restrict__ dst,
                              int* __restrict__ outdeg, int* __restrict__ indeg, int E) {
    int i = blockIdx.x * blockDim.x + threadIdx.x;
    if (i < E) {
        atomicAdd(&outdeg[src[i]], 1);
        atomicAdd(&indeg[dst[i]], 1);
    }
}

// ---------------------------------------------------------------------------
// K2: h = ((feat * outdeg^-1/2) @ W) * weight_type[type_info]  via fp32 WMMA.
// One wave computes a 16x64 tile: 4 accumulators (16x16 f32 = v8f each),
// 16 k-chunks of V_WMMA_F32_16X16X4_F32 (K=64).
//
// A 16x4 f32 layout: lanes 0-15 -> M=lane, K={kc*4+0, kc*4+1};
//                    lanes 16-31 -> M=lane-16, K={kc*4+2, kc*4+3}.
// B 4x16 f32 layout: lanes 0-15 -> N=lane, K={kc*4+0, kc*4+1};
//                    lanes 16-31 -> N=lane-16, K={kc*4+2, kc*4+3}.
// C/D 16x16 f32: vgpr j, lanes 0-15 -> M=j; lanes 16-31 -> M=j+8; N=lane&15.
// ---------------------------------------------------------------------------
__global__ void gemm_wmma_kernel(const float* __restrict__ feat,
                                 const float* __restrict__ weight,
                                 const float* __restrict__ wtype,
                                 const int* __restrict__ type_info,
                                 const int* __restrict__ outdeg,
                                 float* __restrict__ h, int N) {
    __shared__ float Wlds[F_DIM * WPAD];        // 64 rows, padded stride
    __shared__ float Flds[8][16 * FPAD];        // per-wave 16x64 feat tile

    const int tid  = threadIdx.x;               // 0..255 (8 waves)
    const int wave = tid >> 5;
    const int lane = tid & 31;

    // Stage weight into LDS (once per block, all waves).
    for (int i = tid; i < F_DIM * F_DIM; i += 256) {
        int r = i >> 6, c = i & 63;
        Wlds[r * WPAD + c] = weight[i];
    }
    __syncthreads();

    const int tile = blockIdx.x * 8 + wave;     // 16-row tile index (wave-uniform)
    if (tile * 16 < N) {
        const int m0 = tile * 16;
        float* lf = &Flds[wave][0];

        // Stage this wave's 16x64 feat tile into LDS, applying outdeg^-1/2.
        #pragma unroll
        for (int it = 0; it < 8; ++it) {
            int idx = lane + it * 32;           // float4 slot 0..255
            int row = idx >> 4;
            int c4  = idx & 15;
            int m   = m0 + row; if (m >= N) m = N - 1;
            float ds = rsqrtf(fmaxf((float)outdeg[m], 1.0f));
            float4 v = ((const float4*)(feat + (long long)m * F_DIM))[c4];
            float* p = lf + row * FPAD + c4 * 4;
            p[0] = v.x * ds; p[1] = v.y * ds; p[2] = v.z * ds; p[3] = v.w * ds;
        }
        // Wave-local LDS RAW ordering (cross-lane): wait for this wave's DS stores.
        asm volatile("s_wait_dscnt 0x0" ::: "memory");

        const v8f vzero = {0.f,0.f,0.f,0.f,0.f,0.f,0.f,0.f};
        v8f acc[4] = {vzero, vzero, vzero, vzero};

        const int half = lane >> 4;             // 0: lanes 0-15, 1: lanes 16-31
        const int arow = lane & 15;             // A: M row / B: N col
        const int koff = half * 2;              // K sub-offset for this half

        #pragma unroll
        for (int kc = 0; kc < 16; ++kc) {
            const int k = kc * 4 + koff;
            v2f a;
            a.x = lf[arow * FPAD + k];
            a.y = lf[arow * FPAD + k + 1];
            #pragma unroll
            for (int n = 0; n < 4; ++n) {
                v2f b;
                b.x = Wlds[k * WPAD + n * 16 + arow];
                b.y = Wlds[(k + 1) * WPAD + n * 16 + arow];
                acc[n] = __builtin_amdgcn_wmma_f32_16x16x4_f32(
                    false, a, false, b, (short)0, acc[n], false, false);
            }
        }

        // Apply type scale and store the 16x64 tile.
        const int mbase = m0 + half * 8;
        #pragma unroll
        for (int j = 0; j < 8; ++j) {
            int m = mbase + j;
            if (m < N) {
                float ts = wtype[type_info[m]];
                #pragma unroll
                for (int n = 0; n < 4; ++n)
                    h[(long long)m * F_DIM + n * 16 + arow] = acc[n][j] * ts;
            }
        }
    }
}

// ---------------------------------------------------------------------------
// K3: edge scatter: out[dst] += h[src].  16 lanes per edge, float4 gather,
// hardware global_atomic_add_f32 scatter (h and out both L2-resident).
// ---------------------------------------------------------------------------
__global__ void edge_scatter_kernel(const float* __restrict__ h,
                                    const int* __restrict__ src,
                                    const int* __restrict__ dst,
                                    float* __restrict__ out, int E) {
    long long gid = (long long)blockIdx.x * blockDim.x + threadIdx.x;
    int e    = (int)(gid >> 4);
    int lane = (int)(gid & 15);
    if (e >= E) return;
    int s = src[e];
    int d = dst[e];
    float4 v = ((const float4*)(h + (long long)s * F_DIM))[lane];
    float* o = out + (long long)d * F_DIM + lane * 4;
    unsafeAtomicAdd(o + 0, v.x);
    unsafeAtomicAdd(o + 1, v.y);
    unsafeAtomicAdd(o + 2, v.z);
    unsafeAtomicAdd(o + 3, v.w);
}

// ---------------------------------------------------------------------------
// K4: out = out * indeg^-1/2 + bias   (float4 per thread)
// ---------------------------------------------------------------------------
__global__ void finalize_kernel(float* __restrict__ out, const int* __restrict__ indeg,
                                const float* __restrict__ bias, int N) {
    int gid = blockIdx.x * blockDim.x + threadIdx.x;
    if (gid < N * 16) {
        int n  = gid >> 4;
        int c4 = gid & 15;
        float s = rsqrtf(fmaxf((float)indeg[n], 1.0f));
        float4 v = ((float4*)out)[gid];
        float4 b = ((const float4*)bias)[c4];
        v.x = v.x * s + b.x;
        v.y = v.y * s + b.y;
        v.z = v.z * s + b.z;
        v.w = v.w * s + b.w;
        ((float4*)out)[gid] = v;
    }
}

extern "C" void kernel_launch(void* const* d_in, const int* in_sizes, int n_in,
                              void* d_out, int out_size, void* d_ws, size_t ws_size,
                              hipStream_t stream) {
    const float* feat      = (const float*)d_in[0];
    const float* weight    = (const float*)d_in[1];
    const float* wtype     = (const float*)d_in[2];
    const float* bias      = (const float*)d_in[3];
    const int*   src       = (const int*)d_in[4];
    const int*   dst       = (const int*)d_in[5];
    const int*   type_info = (const int*)d_in[6];

    const int E = in_sizes[4];
    const int N = in_sizes[6];

    float* out    = (float*)d_out;
    int*   outdeg = (int*)d_ws;
    int*   indeg  = outdeg + N;
    float* h      = (float*)d_ws + 2 * (size_t)N;   // N*64 floats

    // K0: zero degrees + output accumulator.
    zero_kernel<<<2048, 256, 0, stream>>>(outdeg, 2LL * N, out, (long long)N * F_DIM);

    // K1: degree histograms.
    degree_kernel<<<(E + 255) / 256, 256, 0, stream>>>(src, dst, outdeg, indeg, E);

    // K2: WMMA GEMM with left-norm + type scale.
    int tiles  = (N + 15) / 16;
    int blocks = (tiles + 7) / 8;
    gemm_wmma_kernel<<<blocks, 256, 0, stream>>>(feat, weight, wtype, type_info,
                                                 outdeg, h, N);

    // K3: edge gather/scatter-add.
    long long scat_threads = (long long)E * 16;
    edge_scatter_kernel<<<(unsigned)((scat_threads + 255) / 256), 256, 0, stream>>>(
        h, src, dst, out, E);

    // K4: right-norm + bias.
    finalize_kernel<<<(N * 16 + 255) / 256, 256, 0, stream>>>(out, indeg, bias, N);
}